// CV_STDPModel_11630771437694
// MI455X (gfx1250) — compile-verified
//
#include <hip/hip_runtime.h>
#include <math.h>

typedef __attribute__((ext_vector_type(2))) float v2f;
typedef __attribute__((ext_vector_type(8))) float v8f;

#define T_STEPS 30

__device__ __forceinline__ int refl(int j, int n) {
  if (j < 0) j = -1 - j;          // numpy 'symmetric': x[-1]->x[0]
  if (j >= n) j = 2 * n - 1 - j;
  return j;
}

// ---------------------------------------------------------------------------
// Kernel 1: DoG preprocess. One block per image. Separable 7-tap Gaussian over
// axis0 (batch, symmetric reflect), axis2 (h), axis3 (w); axis1 (C=1) is
// identity. Writes d = |g - x| and atomicMax of the global max (values >= 0,
// so uint compare is order-correct).
// ---------------------------------------------------------------------------
__global__ void __launch_bounds__(256) dog_kernel(const float* __restrict__ x,
                                                  float* __restrict__ dbuf,
                                                  unsigned int* __restrict__ gmax,
                                                  int nimg) {
  __shared__ float aL[784];
  __shared__ float bL[784];
  __shared__ float red[256];
  const int img = blockIdx.x;
  const int tid = threadIdx.x;

  float wk[7];
  float wsum = 0.f;
#pragma unroll
  for (int k = 0; k < 7; ++k) {
    float t = (float)(k - 3) * 0.5f;          // i / SIGMA, SIGMA = 2
    wk[k] = expf(-0.5f * t * t);
    wsum += wk[k];
  }
#pragma unroll
  for (int k = 0; k < 7; ++k) wk[k] /= wsum;

  // axis 0: filter across batch
  for (int pix = tid; pix < 784; pix += 256) {
    float acc = 0.f;
#pragma unroll
    for (int k = 0; k < 7; ++k) {
      int j = refl(img - 3 + k, nimg);
      acc += wk[k] * x[(size_t)j * 784 + pix];
    }
    aL[pix] = acc;
  }
  __syncthreads();

  // axis 2: h
  for (int pix = tid; pix < 784; pix += 256) {
    int h = pix / 28, w = pix - h * 28;
    float acc = 0.f;
#pragma unroll
    for (int k = 0; k < 7; ++k) acc += wk[k] * aL[refl(h - 3 + k, 28) * 28 + w];
    bL[pix] = acc;
  }
  __syncthreads();

  // axis 3: w, then |g - x| and local max
  float lmax = 0.f;
  for (int pix = tid; pix < 784; pix += 256) {
    int h = pix / 28, w = pix - h * 28;
    float acc = 0.f;
#pragma unroll
    for (int k = 0; k < 7; ++k) acc += wk[k] * bL[h * 28 + refl(w - 3 + k, 28)];
    float dv = fabsf(acc - x[(size_t)img * 784 + pix]);
    dbuf[(size_t)img * 784 + pix] = dv;
    lmax = fmaxf(lmax, dv);
  }
  red[tid] = lmax;
  __syncthreads();
  for (int s = 128; s > 0; s >>= 1) {
    if (tid < s) red[tid] = fmaxf(red[tid], red[tid + s]);
    __syncthreads();
  }
  if (tid == 0) atomicMax(gmax, __float_as_uint(red[0]));
}

// ---------------------------------------------------------------------------
// Kernel 2: 30-step SNN. One block (8 wave32) per image. Membrane potentials
// v2/v3 carried in WMMA accumulator registers across the whole time loop
// (C operand of V_WMMA_F32_16X16X4_F32 = old potential -> D = v + conv; exact
// f32 math preserves spike thresholds).
//
// Predication-free design:
//   * A/C operands: unconditional clamped-index LDS loads (invalid K taps are
//     zeroed once on the B side; invalid M/N outputs are never consumed).
//   * Spike stores: destination buffers padded so ALL 32 lanes always have a
//     unique in-bounds slot (pad rows/cols are never read) -> plain ds_store,
//     no exec masking.
//   * Pools/layer1: fixed trip counts with clamped source indices into padded
//     destinations.
//
// Operand layouts (wave32):
//   A 16x4:  lane<16 -> M=lane, {x:K=0, y:K=1}; lane>=16 -> M=lane-16, {K=2,K=3}
//   B 4x16:  lane<16 -> N=lane, {x:K=0, y:K=1}; lane>=16 -> N=lane-16, {K=2,K=3}
//   C/D:     elem r: lane<16 -> (M=r, N=lane); lane>=16 -> (M=r+8, N=lane-16)
// ---------------------------------------------------------------------------
__global__ void __launch_bounds__(256) snn_kernel(const float* __restrict__ dbuf,
                                                  const unsigned int* __restrict__ gmaxp,
                                                  const float* __restrict__ w1,
                                                  const float* __restrict__ w2,
                                                  const float* __restrict__ w3,
                                                  float* __restrict__ out) {
  __shared__ float dL[1024];        // 784 valid + zero pad
  __shared__ float s1L[1024];       // 784 valid + write pad
  __shared__ float s2P[16 * 688];   // conv1 spikes, all-lane padded store
  __shared__ float p1L[768];        // 676 valid + write pad
  __shared__ float s3P[16 * 128];   // conv2 spikes, all-lane padded store
  __shared__ float p2L[256];        // 144 valid + write pad
  __shared__ float v4L[16];
  __shared__ float predL[16];
  __shared__ float w3L[36];

  const int img = blockIdx.x;
  const int tid = threadIdx.x;
  const int lane = tid & 31;
  const int wv = tid >> 5;          // wave id, 0..7
  const int n16 = lane & 15;
  const int half = lane >> 4;
  const int occ = (n16 < 4) ? n16 : 0;   // clamped out-channel for weight loads

  const float gmax = __uint_as_float(*gmaxp);

  // d (pad region = 0 so pad membranes never fire)
#pragma unroll
  for (int j = 0; j < 4; ++j) {
    int p = tid + 256 * j;
    float v = dbuf[(size_t)img * 784 + ((p < 784) ? p : 783)] / gmax;
    dL[p] = (p < 784) ? v : 0.f;
  }
  if (tid < 16) { v4L[tid] = 0.f; predL[tid] = 0.f; }
  if (tid < 36) w3L[tid] = w3[tid];

  // ---- B operands (weights), resident in registers for all 30 steps ----
  v2f B1[3];
#pragma unroll
  for (int c = 0; c < 3; ++c) {
    int k0 = 4 * c + 2 * half;           // taps 0..8; k>=9 zeroed here
    B1[c].x = (n16 < 4 && k0 < 9) ? w1[occ * 9 + ((k0 < 9) ? k0 : 0)] : 0.f;
    B1[c].y = (n16 < 4 && k0 + 1 < 9) ? w1[occ * 9 + ((k0 + 1 < 9) ? k0 + 1 : 0)] : 0.f;
  }
  v2f B2[9];
#pragma unroll
  for (int c = 0; c < 9; ++c) {
    int k0 = 4 * c + 2 * half;           // K = ic*9 + tap, 36 = 9*4 exactly
    B2[c].x = (n16 < 4) ? w2[occ * 36 + k0] : 0.f;
    B2[c].y = (n16 < 4) ? w2[occ * 36 + k0 + 1] : 0.f;
  }

  // ---- loop-invariant per-lane indices ----
  int a1o[6];                            // conv1 im2col tap offsets (clamped)
#pragma unroll
  for (int j = 0; j < 6; ++j) {
    int k = 4 * (j >> 1) + 2 * half + (j & 1);
    int kk = (k < 9) ? k : 8;
    a1o[j] = (kk / 3) * 28 + (kk % 3);
  }
  const int sb1 = n16 * 688 + 8 * half;  // conv1 padded spike-store base

  const int P2 = wv * 16;                // conv2: wave wv owns tile wv
  int a2i[18];
  {
    int pa = P2 + n16;
    int pc = (pa < 121) ? pa : 120;      // clamp; invalid rows never consumed
    int oy = pc / 11, ox = pc - oy * 11;
#pragma unroll
    for (int j = 0; j < 18; ++j) {
      int k = 4 * (j >> 1) + 2 * half + (j & 1);  // 0..35
      int ic = k / 9, tt = k - ic * 9;
      a2i[j] = ic * 169 + (oy + tt / 3) * 13 + (ox + tt % 3);
    }
  }
  const int sb2 = n16 * 128 + 8 * half;  // conv2 padded spike-store base

  // ---- persistent state in registers ----
  float v1r[4] = {0.f, 0.f, 0.f, 0.f};   // layer-1 membrane (4 px/thread)
  v8f vacc1[6];                          // conv1 membranes, up to 6 tiles/wave
#pragma unroll
  for (int i = 0; i < 6; ++i)
#pragma unroll
    for (int r = 0; r < 8; ++r) vacc1[i][r] = 0.f;
  v8f vacc2;                             // conv2 membrane, 1 tile/wave
#pragma unroll
  for (int r = 0; r < 8; ++r) vacc2[r] = 0.f;

  __syncthreads();

#pragma unroll 1
  for (int t = 0; t < T_STEPS; ++t) {
    // ---- layer 1: IF(4.5); fully unconditional over padded 1024 ----
#pragma unroll
    for (int j = 0; j < 4; ++j) {
      int p = tid + 256 * j;
      float v = v1r[j] + dL[p];
      float s = (v >= 4.5f) ? 1.f : 0.f;
      v1r[j] = (s > 0.f) ? 0.f : v;
      s1L[p] = s;
    }
    __syncthreads();

    // ---- conv1 (1->4, 26x26) + IF(4.5): 43 tiles over 8 waves ----
#pragma unroll
    for (int i = 0; i < 6; ++i) {
      int tile = wv + 8 * i;
      if (tile < 43) {                   // wave-uniform at runtime
        int P = tile * 16;
        int pa = P + n16;
        int pc = (pa < 676) ? pa : 675;  // clamp row; pad rows never consumed
        int abase = pc + 2 * (pc / 26);  // oy*28 + ox  (max 725; +ofs <= 783)
        v2f A0, A1, A2;
        A0.x = s1L[abase + a1o[0]]; A0.y = s1L[abase + a1o[1]];
        A1.x = s1L[abase + a1o[2]]; A1.y = s1L[abase + a1o[3]];
        A2.x = s1L[abase + a1o[4]]; A2.y = s1L[abase + a1o[5]];
        v8f acc = vacc1[i];
        acc = __builtin_amdgcn_wmma_f32_16x16x4_f32(false, A0, false, B1[0],
                                                    (short)0, acc, false, false);
        acc = __builtin_amdgcn_wmma_f32_16x16x4_f32(false, A1, false, B1[1],
                                                    (short)0, acc, false, false);
        acc = __builtin_amdgcn_wmma_f32_16x16x4_f32(false, A2, false, B1[2],
                                                    (short)0, acc, false, false);
#pragma unroll
        for (int r = 0; r < 8; ++r) {
          float v = acc[r];
          float s = (v >= 4.5f) ? 1.f : 0.f;
          acc[r] = (s > 0.f) ? 0.f : v;
          s2P[sb1 + P + r] = s;          // unconditional padded store
        }
        vacc1[i] = acc;
      }
    }
    __syncthreads();

    // ---- maxpool 2x2: 26x26 -> 13x13; 3 fixed unconditional iterations ----
#pragma unroll
    for (int j = 0; j < 3; ++j) {
      int idx = tid + 256 * j;           // 0..767 into padded p1L
      int isrc = (idx < 676) ? idx : 675;
      int oc = isrc / 169, rem = isrc - oc * 169;
      int y = rem / 13, xx = rem - y * 13;
      const float* sp = &s2P[oc * 688 + (2 * y) * 26 + 2 * xx];
      p1L[idx] = fmaxf(fmaxf(sp[0], sp[1]), fmaxf(sp[26], sp[27]));
    }
    __syncthreads();

    // ---- conv2 (4->4, 11x11) + IF(1.0): 8 tiles = 8 waves ----
    {
      v2f A;
      v8f acc = vacc2;
#pragma unroll
      for (int c = 0; c < 9; ++c) {
        A.x = p1L[a2i[2 * c]];
        A.y = p1L[a2i[2 * c + 1]];
        acc = __builtin_amdgcn_wmma_f32_16x16x4_f32(false, A, false, B2[c],
                                                    (short)0, acc, false, false);
      }
#pragma unroll
      for (int r = 0; r < 8; ++r) {
        float v = acc[r];
        float s = (v >= 1.0f) ? 1.f : 0.f;
        acc[r] = (s > 0.f) ? 0.f : v;
        s3P[sb2 + P2 + r] = s;           // unconditional padded store
      }
      vacc2 = acc;
    }
    __syncthreads();

    // ---- maxpool 2x2 pad=1: 11x11 -> 6x6; one unconditional store/thread ----
    {
      int idx = (tid < 144) ? tid : 143; // clamped source, padded dest
      int oc = idx / 36, rem = idx - oc * 36;
      int y = rem / 6, xx = rem - y * 6;
      int r0 = 2 * y - 1, c0 = 2 * xx - 1;
      float m = 0.f;
#pragma unroll
      for (int dy = 0; dy < 2; ++dy)
#pragma unroll
        for (int dx = 0; dx < 2; ++dx) {
          int rr = r0 + dy, cc = c0 + dx;
          // clamped read; contribution masked (spikes >= 0 so OOB=0 is exact)
          int rrc = (rr < 0) ? 0 : ((rr > 10) ? 10 : rr);
          int ccc = (cc < 0) ? 0 : ((cc > 10) ? 10 : cc);
          float vv = s3P[oc * 128 + rrc * 11 + ccc];
          bool inb = (rr >= 0) && (rr < 11) && (cc >= 0) && (cc < 11);
          m = fmaxf(m, inb ? vv : 0.f);
        }
      p2L[tid] = m;                      // threads >=144 fill pad slots
    }
    __syncthreads();

    // ---- conv3 (4->1, 4x4) + LIF(tau=2): 576 MACs, scalar on 16 lanes ----
    if (tid < 16) {
      int oy = tid >> 2, ox = tid & 3;
      float c3 = 0.f;
#pragma unroll
      for (int ic = 0; ic < 4; ++ic)
#pragma unroll
        for (int ty = 0; ty < 3; ++ty)
#pragma unroll
          for (int tx = 0; tx < 3; ++tx)
            c3 += w3L[ic * 9 + ty * 3 + tx] * p2L[ic * 36 + (oy + ty) * 6 + (ox + tx)];
      float v = v4L[tid];
      v = v + (c3 - v) * 0.5f;           // exact: /2 == *0.5
      float s = (v >= 1.0f) ? 1.f : 0.f;
      v4L[tid] = (s > 0.f) ? 0.f : v;
      predL[tid] += s;
    }
    __syncthreads();
  }

  if (tid < 16) out[(size_t)img * 16 + tid] = predL[tid] / 30.0f;
}

// ---------------------------------------------------------------------------
// Launch: ws layout = [gmax u32 @0 .. pad .. d (nimg*784 f32) @256].
// ---------------------------------------------------------------------------
extern "C" void kernel_launch(void* const* d_in, const int* in_sizes, int n_in,
                              void* d_out, int out_size, void* d_ws, size_t ws_size,
                              hipStream_t stream) {
  const float* x  = (const float*)d_in[0];
  const float* w1 = (const float*)d_in[1];
  const float* w2 = (const float*)d_in[2];
  const float* w3 = (const float*)d_in[3];
  float* out = (float*)d_out;
  const int nimg = in_sizes[0] / 784;

  unsigned int* gmax = (unsigned int*)d_ws;
  float* dbuf = (float*)((char*)d_ws + 256);

  hipMemsetAsync(gmax, 0, sizeof(unsigned int), stream);   // graph-capturable
  dog_kernel<<<nimg, 256, 0, stream>>>(x, dbuf, gmax, nimg);
  snn_kernel<<<nimg, 256, 0, stream>>>(dbuf, gmax, w1, w2, w3, out);
  (void)n_in; (void)out_size; (void)ws_size;
}